// ConcatAttention_16956530884977
// MI455X (gfx1250) — compile-verified
//
#include <hip/hip_runtime.h>
#include <hip/hip_bf16.h>
#include <stdint.h>

// Problem dims (fixed by the reference setup)
constexpr int kB  = 2;
constexpr int kTq = 512;
constexpr int kTk = 1024;
constexpr int kH  = 128;

typedef float v2f __attribute__((ext_vector_type(2)));
typedef float v8f __attribute__((ext_vector_type(8)));

// ---------------------------------------------------------------------------
// Hardware tanh: gfx1250 has V_TANH_F32 (trans op). Prefer the builtin (gets
// correct hazard/delay handling from the compiler); fall back to inline asm
// with a trailing v_nop to cover the 1-cycle TRANS32 dependency gap.
// ---------------------------------------------------------------------------
__device__ __forceinline__ float fast_tanh(float x) {
#if __has_builtin(__builtin_amdgcn_tanhf)
    return __builtin_amdgcn_tanhf(x);
#else
    float r;
    asm("v_tanh_f32 %0, %1\n\tv_nop" : "=v"(r) : "v"(x));
    return r;
#endif
}

__device__ __forceinline__ float wave_reduce_max(float v) {
#pragma unroll
    for (int off = 16; off > 0; off >>= 1)
        v = fmaxf(v, __shfl_xor(v, off, 32));
    return v;
}

__device__ __forceinline__ float wave_reduce_sum(float v) {
#pragma unroll
    for (int off = 16; off > 0; off >>= 1)
        v += __shfl_xor(v, off, 32);
    return v;
}

// ---------------------------------------------------------------------------
// Kernel 1: Y(M,128) = X(M,128) @ W(128,128), row-major, exact fp32 via
// V_WMMA_F32_16X16X4_F32. One wave per 16x16 output tile, 4 waves per block.
//
// ISA fragment layouts (wave32):
//   A 16x4 f32: lane L<16 -> {A[M=L][K=0], A[M=L][K=1]},
//               lane L>=16 -> {A[M=L-16][K=2], A[M=L-16][K=3]}
//   B 4x16 f32: lane L<16 -> {B[K=0][N=L], B[K=1][N=L]},
//               lane L>=16 -> {B[K=2][N=L-16], B[K=3][N=L-16]}
//   C/D 16x16:  VGPR v, lane L<16 -> D[M=v][N=L]; lane>=16 -> D[M=v+8][N=L-16]
// ---------------------------------------------------------------------------
__global__ void gemm_wmma_f32(const float* __restrict__ X,
                              const float* __restrict__ W,
                              float* __restrict__ Y, int M) {
    const int lane = threadIdx.x & 31;
    const int wave = threadIdx.x >> 5;
    const int tile = blockIdx.x * 4 + wave;

    const int tiles_n = kH / 16;              // 8
    const int tm = tile / tiles_n;
    const int tn = tile % tiles_n;
    const int m0 = tm * 16;
    const int n0 = tn * 16;
    if (m0 >= M) return;                      // whole wave exits together

    const int lr = lane & 15;                 // A row / B col within tile
    const int hi = lane >> 4;                 // 0: K=0,1  1: K=2,3

    const float* xrow = X + (size_t)(m0 + lr) * kH;
    v8f c = {0.f, 0.f, 0.f, 0.f, 0.f, 0.f, 0.f, 0.f};

    for (int kk = 0; kk < kH; kk += 4) {
        const int ka = kk + hi * 2;
        v2f a = *(const v2f*)(xrow + ka);     // contiguous, 8B aligned
        v2f b;
        b.x = W[(size_t)(ka + 0) * kH + n0 + lr];
        b.y = W[(size_t)(ka + 1) * kH + n0 + lr];
        c = __builtin_amdgcn_wmma_f32_16x16x4_f32(false, a, false, b,
                                                  (short)0, c, false, false);
    }

    const int col = n0 + lr;
    const int rowbase = m0 + hi * 8;
#pragma unroll
    for (int v = 0; v < 8; ++v)
        Y[(size_t)(rowbase + v) * kH + col] = c[v];
}

// ---------------------------------------------------------------------------
// Kernel 2: scores + masked softmax (the VALU-heavy part).
// One 256-thread block per (b,t) row; each thread owns 4 key positions.
//   score[s] = sum_h scale[h] * tanh(q[b,t,h] + k[b,s,h]) - 1e9*(1-mask[b,s])
// then softmax over s, written to attn (B, Tq, Tk).
// ---------------------------------------------------------------------------
__global__ void scores_softmax(const float* __restrict__ qproj,  // (B*Tq, H)
                               const float* __restrict__ kproj,  // (B*Tk, H)
                               const float* __restrict__ scale,  // (H)
                               const uint8_t* __restrict__ mask, // (B, Tk)
                               float* __restrict__ attn) {       // (B*Tq, Tk)
    __shared__ float s_q[kH];
    __shared__ float s_sc[kH];
    __shared__ float s_red[8];

    const int row = blockIdx.x;        // b*Tq + t
    const int b   = row / kTq;
    const int tid = threadIdx.x;

    if (tid < kH) {
        s_q[tid]  = qproj[(size_t)row * kH + tid];
        s_sc[tid] = scale[tid];
    }
    __syncthreads();

    const float*   kb = kproj + (size_t)b * kTk * kH;
    const uint8_t* mb = mask + (size_t)b * kTk;

    float sc[4];
#pragma unroll
    for (int i = 0; i < 4; ++i) {
        const int s = tid + i * 256;
        const float* kr = kb + (size_t)s * kH;
        float acc = 0.f;
#pragma unroll 4
        for (int h = 0; h < kH; h += 4) {
            float4 kv = *(const float4*)(kr + h);
            acc = fmaf(s_sc[h + 0], fast_tanh(s_q[h + 0] + kv.x), acc);
            acc = fmaf(s_sc[h + 1], fast_tanh(s_q[h + 1] + kv.y), acc);
            acc = fmaf(s_sc[h + 2], fast_tanh(s_q[h + 2] + kv.z), acc);
            acc = fmaf(s_sc[h + 3], fast_tanh(s_q[h + 3] + kv.w), acc);
        }
        acc += -1e9f * (1.0f - (float)mb[s]);
        sc[i] = acc;
    }

    // --- block softmax over 1024 values (8 wave32 waves) ---
    const int wid = tid >> 5, lid = tid & 31;

    float m = fmaxf(fmaxf(sc[0], sc[1]), fmaxf(sc[2], sc[3]));
    m = wave_reduce_max(m);
    if (lid == 0) s_red[wid] = m;
    __syncthreads();
    float bmax = s_red[0];
#pragma unroll
    for (int w = 1; w < 8; ++w) bmax = fmaxf(bmax, s_red[w]);
    __syncthreads();  // s_red reused below

    float e[4], psum = 0.f;
#pragma unroll
    for (int i = 0; i < 4; ++i) {
        e[i] = __expf(sc[i] - bmax);
        psum += e[i];
    }
    psum = wave_reduce_sum(psum);
    if (lid == 0) s_red[wid] = psum;
    __syncthreads();
    float bsum = 0.f;
#pragma unroll
    for (int w = 0; w < 8; ++w) bsum += s_red[w];
    const float inv = 1.0f / bsum;

    float* arow = attn + (size_t)row * kTk;
#pragma unroll
    for (int i = 0; i < 4; ++i)
        arow[tid + i * 256] = e[i] * inv;
}

// ---------------------------------------------------------------------------
// Kernel 3: context[b] = attn[b] (Tq x Tk) @ value[b] (Tk x H), fp32 WMMA.
// One wave per 16x16 output tile; K loop over Tk in steps of 4.
// ---------------------------------------------------------------------------
__global__ void context_wmma_f32(const float* __restrict__ attn,
                                 const float* __restrict__ value,
                                 float* __restrict__ ctx) {
    const int lane = threadIdx.x & 31;
    const int wave = threadIdx.x >> 5;
    const int gtile = blockIdx.x * 4 + wave;

    const int tiles_n = kH / 16;                       // 8
    const int tiles_per_b = (kTq / 16) * tiles_n;      // 256
    const int b    = gtile / tiles_per_b;
    const int tile = gtile % tiles_per_b;
    const int m0 = (tile / tiles_n) * 16;
    const int n0 = (tile % tiles_n) * 16;

    const int lr = lane & 15;
    const int hi = lane >> 4;

    const float* arow = attn + (size_t)b * kTq * kTk + (size_t)(m0 + lr) * kTk;
    const float* vb   = value + (size_t)b * kTk * kH;

    v8f c = {0.f, 0.f, 0.f, 0.f, 0.f, 0.f, 0.f, 0.f};
    for (int kk = 0; kk < kTk; kk += 4) {
        const int ka = kk + hi * 2;
        v2f a = *(const v2f*)(arow + ka);
        v2f bf;
        bf.x = vb[(size_t)(ka + 0) * kH + n0 + lr];
        bf.y = vb[(size_t)(ka + 1) * kH + n0 + lr];
        c = __builtin_amdgcn_wmma_f32_16x16x4_f32(false, a, false, bf,
                                                  (short)0, c, false, false);
    }

    float* crow = ctx + (size_t)b * kTq * kH;
    const int col = n0 + lr;
    const int rowbase = m0 + hi * 8;
#pragma unroll
    for (int v = 0; v < 8; ++v)
        crow[(size_t)(rowbase + v) * kH + col] = c[v];
}

// ---------------------------------------------------------------------------
extern "C" void kernel_launch(void* const* d_in, const int* in_sizes, int n_in,
                              void* d_out, int out_size, void* d_ws, size_t ws_size,
                              hipStream_t stream) {
    const float*   query = (const float*)d_in[0];   // (B, Tq, H)
    const float*   value = (const float*)d_in[1];   // (B, Tk, H)
    const uint8_t* mask  = (const uint8_t*)d_in[2]; // (B, Tk) bool
    const float*   W1    = (const float*)d_in[3];   // (H, H)
    const float*   W2    = (const float*)d_in[4];   // (H, H)
    const float*   scale = (const float*)d_in[5];   // (H)

    // Output tuple order: context (B,Tq,H) then attention_weights (B,Tq,Tk)
    float* ctx  = (float*)d_out;
    float* attn = (float*)d_out + (size_t)kB * kTq * kH;

    // Workspace: q projection (512 KB) + k projection (1 MB)
    float* qproj = (float*)d_ws;
    float* kproj = qproj + (size_t)kB * kTq * kH;

    // q = query @ W1 : M = 2048/2 = 1024 rows -> 512 tiles -> 128 blocks x 4 waves
    gemm_wmma_f32<<<128, 128, 0, stream>>>(query, W1, qproj, kB * kTq);
    // k = value @ W2 : M = 2048 rows -> 1024 tiles -> 256 blocks
    gemm_wmma_f32<<<256, 128, 0, stream>>>(value, W2, kproj, kB * kTk);
    // scores + masked softmax: one block per (b,t) row
    scores_softmax<<<kB * kTq, 256, 0, stream>>>(qproj, kproj, scale, mask, attn);
    // context = attn @ value : 512 tiles -> 128 blocks
    context_wmma_f32<<<128, 128, 0, stream>>>(attn, value, ctx);
}